// EMAFixed_23338852286472
// MI455X (gfx1250) — compile-verified
//
#include <hip/hip_runtime.h>

typedef float v2f __attribute__((ext_vector_type(2)));
typedef float v8f __attribute__((ext_vector_type(8)));

#define NB   8
#define NC   512
#define NT   16384
#define WAVES_PER_BLOCK 8

// integer power by repeated multiply (loop-invariant use only)
__device__ __forceinline__ float gpow_int(float g, int e) {
    float p = 1.0f;
    for (int i = 0; i < e; ++i) p *= g;
    return p;
}

__global__ __launch_bounds__(WAVES_PER_BLOCK * 32)
void ema_scan_wmma(const float* __restrict__ x,
                   const float* __restrict__ weight,
                   float* __restrict__ y)
{
    const int lane = threadIdx.x & 31;
    const int wave = threadIdx.x >> 5;
    const int row  = blockIdx.x * WAVES_PER_BLOCK + wave;   // 0..4095 = b*NC + c
    const int c    = row & (NC - 1);
    const long long base = (long long)row * NT;

    const float g = weight[c];
    const float s = 1.0f - g;

    const int col  = lane & 15;   // WMMA column (chunk within tile); M row t for A
    const int half = lane >> 4;

    // ---- loop-invariant per row ------------------------------------------
    // A fragments: L[t][k] = (k<=t) ? s*g^(t-k) : 0   (u-scale folded into A)
    // A 16x4 f32 layout: lanes hold M = lane&15; VGPRs = K pair (2*half)+0/1
    v2f afrag[4];
#pragma unroll
    for (int kk = 0; kk < 4; ++kk) {
        const int k0 = 4 * kk + 2 * half;
        const int t  = col;
        afrag[kk].x = (k0     <= t) ? s * gpow_int(g, t - k0)     : 0.0f;
        afrag[kk].y = (k0 + 1 <= t) ? s * gpow_int(g, t - k0 - 1) : 0.0f;
    }

    const float g2 = g * g, g4 = g2 * g2, g8 = g4 * g4, g16 = g8 * g8;
    // g^(t+1) fixup factors, t = v + 8*half (same for both tiles)
    float fpow[8];
    fpow[0] = half ? (g8 * g) : g;
#pragma unroll
    for (int v = 1; v < 8; ++v) fpow[v] = fpow[v - 1] * g;
    // Kogge-Stone weights over 32 chunks: h = g^16
    const float h1 = g16, h2 = h1 * h1, h4 = h2 * h2, h8 = h4 * h4, h16 = h8 * h8;
    const float g512 = h16 * h16;               // decay across one 512-elem iteration
    const float hj   = gpow_int(g16, lane);     // h^lane (chunk index = lane)

    // ---- stream the row: 32 iterations of 512 elements -------------------
    float carry = 0.0f;
    for (int blk = 0; blk < NT / 512; ++blk) {
        const float* xb = x + base + blk * 512;

        // prefetch next 2KB iteration into near caches
        __builtin_prefetch(xb + 512 + lane * 16, 0, 3);

        // B fragments: tile A = chunks 0..15, tile B = chunks 16..31
        v2f bfragA[4], bfragB[4];
#pragma unroll
        for (int kk = 0; kk < 4; ++kk) {
            const int off = col * 16 + 4 * kk + 2 * half;
            bfragA[kk] = *(const v2f*)(xb + off);
            bfragB[kk] = *(const v2f*)(xb + 256 + off);
        }

        // local scans: 32 chunks at once via two accumulator sets
        v8f accA = {}, accB = {};
#pragma unroll
        for (int kk = 0; kk < 4; ++kk) {
            accA = __builtin_amdgcn_wmma_f32_16x16x4_f32(
                false, afrag[kk], false, bfragA[kk], (short)0, accA, false, false);
            accB = __builtin_amdgcn_wmma_f32_16x16x4_f32(
                false, afrag[kk], false, bfragB[kk], (short)0, accB, false, false);
        }

        // gather chunk-final values s_l onto lane l (l = chunk index 0..31):
        // t=15 rows live on lanes 16..31 (acc[7]); tile B's are already in place
        float sA = __shfl(accA[7], 16 + col, 32);
        float v  = half ? accB[7] : sA;

        // 32-wide weighted inclusive scan: v_l = sum_{i<=l} h^(l-i) s_i
        { float o = __shfl(v, (lane - 1)  & 31, 32); v += h1  * ((lane >= 1)  ? o : 0.0f); }
        { float o = __shfl(v, (lane - 2)  & 31, 32); v += h2  * ((lane >= 2)  ? o : 0.0f); }
        { float o = __shfl(v, (lane - 4)  & 31, 32); v += h4  * ((lane >= 4)  ? o : 0.0f); }
        { float o = __shfl(v, (lane - 8)  & 31, 32); v += h8  * ((lane >= 8)  ? o : 0.0f); }
        { float o = __shfl(v, (lane - 16) & 31, 32); v += h16 * ((lane >= 16) ? o : 0.0f); }

        // carry entering chunk l: c_l = h^l * carry + v_{l-1}
        float e = __shfl(v, (lane - 1) & 31, 32);
        e = (lane >= 1) ? e : 0.0f;
        const float cl = hj * carry + e;

        // broadcast per-column carries for each tile (independent bpermutes)
        const float cA  = __shfl(cl, col, 32);
        const float cB  = __shfl(cl, 16 + col, 32);
        const float v31 = __shfl(v, 31, 32);

        // final values + coalesced stores: lane writes 8 contiguous floats/tile
        float4 lo, hi;
        lo.x = accA[0] + fpow[0] * cA;  lo.y = accA[1] + fpow[1] * cA;
        lo.z = accA[2] + fpow[2] * cA;  lo.w = accA[3] + fpow[3] * cA;
        hi.x = accA[4] + fpow[4] * cA;  hi.y = accA[5] + fpow[5] * cA;
        hi.z = accA[6] + fpow[6] * cA;  hi.w = accA[7] + fpow[7] * cA;

        float* yb = y + base + blk * 512 + col * 16 + 8 * half;
        *(float4*)(yb)     = lo;
        *(float4*)(yb + 4) = hi;

        lo.x = accB[0] + fpow[0] * cB;  lo.y = accB[1] + fpow[1] * cB;
        lo.z = accB[2] + fpow[2] * cB;  lo.w = accB[3] + fpow[3] * cB;
        hi.x = accB[4] + fpow[4] * cB;  hi.y = accB[5] + fpow[5] * cB;
        hi.z = accB[6] + fpow[6] * cB;  hi.w = accB[7] + fpow[7] * cB;

        *(float4*)(yb + 256) = lo;
        *(float4*)(yb + 260) = hi;

        // iteration-level carry: value at last element of the 512-block
        carry = g512 * carry + v31;
    }
}

extern "C" void kernel_launch(void* const* d_in, const int* in_sizes, int n_in,
                              void* d_out, int out_size, void* d_ws, size_t ws_size,
                              hipStream_t stream) {
    const float* x = (const float*)d_in[0];
    const float* w = (const float*)d_in[1];
    float*       y = (float*)d_out;

    dim3 block(WAVES_PER_BLOCK * 32);            // 256 threads = 8 waves
    dim3 grid((NB * NC) / WAVES_PER_BLOCK);      // 512 blocks -> 4096 rows
    ema_scan_wmma<<<grid, block, 0, stream>>>(x, w, y);
}